// QRDQNNStepTDError_30073361007339
// MI455X (gfx1250) — compile-verified
//
#include <hip/hip_runtime.h>

// ---------------------------------------------------------------------------
// QRDQN n-step TD error for MI455X (gfx1250, wave32).
//   B=16384, N=32, TAU=64, T=5, gamma=0.99 (sizes derived from in_sizes).
//
// Roofline: gathers touch only ~8.4 MB (0.4 us @ 23.3 TB/s); the B*TAU^2 =
// 67M pairwise quantile-Huber terms at 7 VALU ops each dominate -> VALU-bound.
// The data-dependent Huber x |tau_hat - indicator| core is not a matmul, so
// matrix units only serve reductions:
//
// Kernel 1: one wave per sample. Coalesced float2 gathers of the action rows,
//           v_readlane (SGPR) broadcast of target quantiles, 4 (i,j) terms
//           per lane per unrolled iteration using the branch-free identity
//           huber(u) = m*(|u| - 0.5*m), m = min(|u|,1). Butterfly reduce.
// Kernel 2: deterministic weighted-mean loss in a single wave; the 32 lane
//           partials collapse through V_WMMA_F32_16X16X4_F32 (all-ones B).
// ---------------------------------------------------------------------------

typedef __attribute__((ext_vector_type(2))) float v2f;
typedef __attribute__((ext_vector_type(8))) float v8f;

#ifndef QR_TAU
#define QR_TAU 64
#endif
#define QR_GAMMA 0.99f
#define WAVES_PER_BLOCK 8

__device__ __forceinline__ float lane_bcast(float v, int srcLane) {
  return __builtin_bit_cast(
      float, __builtin_amdgcn_readlane(__builtin_bit_cast(int, v), srcLane));
}

// One pairwise quantile-Huber term, 7 VALU ops:
//   u = tj - si; m = min(|u|,1); huber = m*(|u| - 0.5m)
//   w = (u<=0) ? 1-th : th;  return huber*w  (caller fma-accumulates m*(t*w))
__device__ __forceinline__ float qh_term(float tj, float si, float th,
                                         float one_m_th) {
  float u  = tj - si;
  float au = __builtin_fabsf(u);
  float m  = __builtin_fminf(au, 1.0f);          // v_min_num_f32 with |u| mod
  float t  = __builtin_fmaf(-0.5f, m, au);       // |u| - 0.5*m
  float w  = (u <= 0.0f) ? one_m_th : th;        // |tau_hat - indicator|
  return m * (t * w);
}

__global__ __launch_bounds__(WAVES_PER_BLOCK * 32)
void qrdqn_td_kernel(const float* __restrict__ q,
                     const float* __restrict__ next_n_q,
                     const int*   __restrict__ action,
                     const int*   __restrict__ next_n_action,
                     const float* __restrict__ reward,
                     const float* __restrict__ done,
                     float*       __restrict__ td,      // d_out + 1, length B
                     int B, int N, int nstep) {
  const int wid  = threadIdx.x >> 5;
  const int lane = threadIdx.x & 31;
  int b = blockIdx.x * WAVES_PER_BLOCK + wid;
  if (b >= B) return;
  b = __builtin_amdgcn_readfirstlane(b);   // wave-uniform -> SMEM scalar loads

  // Gather the chosen-action rows (64 floats each, 2 per lane, coalesced b64).
  const float2* qrow = reinterpret_cast<const float2*>(
      q + ((size_t)b * N + action[b]) * QR_TAU);
  const float2* trow = reinterpret_cast<const float2*>(
      next_n_q + ((size_t)b * N + next_n_action[b]) * QR_TAU);
  const float2 s = qrow[lane];   // q quantiles i = 2*lane, 2*lane+1
  const float2 t = trow[lane];   // target quantiles j = 2*lane, 2*lane+1

  // n-step discounted reward and value_gamma = gamma^nstep (wave-uniform).
  float r = 0.0f, g = 1.0f;
  for (int k = 0; k < nstep; ++k) {
    r = __builtin_fmaf(g, reward[(size_t)k * B + b], r);
    g *= QR_GAMMA;
  }
  const float nd = (1.0f - done[b]) * g;
  const float tx = __builtin_fmaf(nd, t.x, r);   // target, even j
  const float ty = __builtin_fmaf(nd, t.y, r);   // target, odd  j

  const float th0  = ((float)(2 * lane) + 0.5f) * (1.0f / QR_TAU);
  const float th1  = th0 + (1.0f / QR_TAU);
  const float oth0 = 1.0f - th0;
  const float oth1 = 1.0f - th1;

  float acc = 0.0f;
  #pragma unroll
  for (int jl = 0; jl < 32; ++jl) {
    const float tjx = lane_bcast(tx, jl);        // j = 2*jl   (SGPR)
    const float tjy = lane_bcast(ty, jl);        // j = 2*jl+1 (SGPR)
    acc += qh_term(tjx, s.x, th0, oth0);
    acc += qh_term(tjy, s.x, th0, oth0);
    acc += qh_term(tjx, s.y, th1, oth1);
    acc += qh_term(tjy, s.y, th1, oth1);
  }

  // Wave32 butterfly: sum over all i rows (sum over j done; mean over i next).
  #pragma unroll
  for (int off = 16; off > 0; off >>= 1) acc += __shfl_xor(acc, off, 32);
  if (lane == 0) td[b] = acc * (1.0f / QR_TAU);
}

// Deterministic loss = mean(td * weight). Single wave; final 32->1 collapse
// uses V_WMMA_F32_16X16X4_F32 with an all-ones B matrix:
//   A[m][0] = p_m (lanes 0-15), A[m][2] = p_{m+16} (lanes 16-31), K=1,3 zero
//   => D[m][*] = p_m + p_{m+16};  lane0 holds rowsums 0..7, lane16 holds 8..15.
__global__ __launch_bounds__(32)
void qrdqn_loss_reduce(const float* __restrict__ td,
                       const float* __restrict__ weight,
                       float*       __restrict__ loss,
                       int B, float invB) {
  const int lane = threadIdx.x;
  float p = 0.0f;
  for (int k = lane; k < B; k += 32)
    p = __builtin_fmaf(td[k], weight[k], p);

  v2f a;  a.x = p;     a.y = 0.0f;
  v2f bm; bm.x = 1.0f; bm.y = 1.0f;
  v8f c = {};
  c = __builtin_amdgcn_wmma_f32_16x16x4_f32(
      /*neg_a=*/false, a, /*neg_b=*/false, bm,
      /*c_mod=*/(short)0, c, /*reuse_a=*/false, /*reuse_b=*/false);

  float s8 = 0.0f;
  #pragma unroll
  for (int v = 0; v < 8; ++v) s8 += c[v];
  const float tot = lane_bcast(s8, 0) + lane_bcast(s8, 16);
  if (lane == 0) loss[0] = tot * invB;
}

extern "C" void kernel_launch(void* const* d_in, const int* in_sizes, int n_in,
                              void* d_out, int out_size, void* d_ws,
                              size_t ws_size, hipStream_t stream) {
  (void)n_in; (void)out_size; (void)d_ws; (void)ws_size;
  const float* q       = (const float*)d_in[0];
  const float* nq      = (const float*)d_in[1];
  const int*   act     = (const int*)d_in[2];
  const int*   nact    = (const int*)d_in[3];
  const float* reward  = (const float*)d_in[4];
  const float* done    = (const float*)d_in[5];
  const float* weight  = (const float*)d_in[6];

  const int B     = in_sizes[2];                // action: (B,)
  const int N     = (in_sizes[0] / B) / QR_TAU; // q: (B, N, TAU)
  const int nstep = in_sizes[4] / B;            // reward: (T, B)

  float* out = (float*)d_out;   // out[0] = loss, out[1..B] = td_err_per_sample
  float* td  = out + 1;

  dim3 block(WAVES_PER_BLOCK * 32);
  dim3 grid((B + WAVES_PER_BLOCK - 1) / WAVES_PER_BLOCK);
  qrdqn_td_kernel<<<grid, block, 0, stream>>>(q, nq, act, nact, reward, done,
                                              td, B, N, nstep);
  qrdqn_loss_reduce<<<dim3(1), dim3(32), 0, stream>>>(td, weight, out, B,
                                                      1.0f / (float)B);
}